// SAGEMeanAgg_11845519802671
// MI455X (gfx1250) — compile-verified
//
#include <hip/hip_runtime.h>
#include <stdint.h>

// SAGE mean-aggregation, CSR-rebuild strategy (avoids 82M f32 atomics):
//  K0: x = dropout(feat_src,u_src) (float4), zero deg
//  K1: deg histogram over dst            (int atomics, 0.64M)
//  K2: exclusive scan (1 block)          -> offsets, cursor
//  K3: scatter src-ids into CSR slots    (int atomic-return, 0.64M)
//  K4: wave-per-dst gather-sum + mean + dropout(h_self) + relu  (no atomics)
// K4 stages gathered rows through LDS with gfx1250 async global->LDS copies
// (double buffered, ASYNCcnt) when the builtin exists; else direct b128 loads
// with global_prefetch_b8 lookahead.

#define D_FEAT 128
#define WAVE 32
#define BLOCK 256
#define WAVES_PER_BLOCK (BLOCK / WAVE)

#if __has_builtin(__builtin_amdgcn_global_load_async_to_lds_b128)
#define HAVE_ASYNC_LDS 1
#endif

#ifdef HAVE_ASYNC_LDS
typedef int v4i __attribute__((ext_vector_type(4)));
typedef __attribute__((address_space(1))) v4i* gv4i_p;  // global int4*
typedef __attribute__((address_space(3))) v4i* lv4i_p;  // LDS int4*
#if __has_builtin(__builtin_amdgcn_s_wait_asynccnt)
#define WAIT_ASYNC(n) __builtin_amdgcn_s_wait_asynccnt(n)
#else
#define WAIT_ASYNC(n) asm volatile("s_wait_asynccnt " #n ::: "memory")
#endif
// One wave moves one 512B row: each lane copies 16 bytes (b128).
__device__ __forceinline__ void async_row_16B(const float* g, float* l) {
  __builtin_amdgcn_global_load_async_to_lds_b128((gv4i_p)g, (lv4i_p)l, 0, 0);
}
#endif

__device__ __forceinline__ float4 dropout4(float4 f, float4 u) {
  float4 r;
  r.x = (u.x < 0.5f) ? f.x * 2.0f : 0.0f;
  r.y = (u.y < 0.5f) ? f.y * 2.0f : 0.0f;
  r.z = (u.z < 0.5f) ? f.z * 2.0f : 0.0f;
  r.w = (u.w < 0.5f) ? f.w * 2.0f : 0.0f;
  return r;
}

__device__ __forceinline__ void acc4(float4& a, const float4 v) {
  a.x += v.x; a.y += v.y; a.z += v.z; a.w += v.w;
}

// ---------------- K0: dropout(feat_src) -> x ; deg = 0 ----------------
__global__ __launch_bounds__(BLOCK)
void sage_init_x_deg(const float* __restrict__ feat, const float* __restrict__ u,
                     float* __restrict__ x, int* __restrict__ deg,
                     int n_vec4, int n_nodes) {
  int tid = blockIdx.x * blockDim.x + threadIdx.x;
  int stride = gridDim.x * blockDim.x;
  const float4* f4 = (const float4*)feat;
  const float4* u4 = (const float4*)u;
  float4* x4 = (float4*)x;
  for (int i = tid; i < n_vec4; i += stride) x4[i] = dropout4(f4[i], u4[i]);
  for (int i = tid; i < n_nodes; i += stride) deg[i] = 0;
}

// ---------------- K1: degree histogram ----------------
__global__ __launch_bounds__(BLOCK)
void sage_hist(const int* __restrict__ dst, int* __restrict__ deg, int n_edges) {
  int tid = blockIdx.x * blockDim.x + threadIdx.x;
  int stride = gridDim.x * blockDim.x;
  for (int e = tid; e < n_edges; e += stride) atomicAdd(&deg[dst[e]], 1);
}

// ---------------- K2: exclusive scan (single block, 1024 thr) ----------------
__global__ __launch_bounds__(1024)
void sage_scan(const int* __restrict__ deg, int* __restrict__ offsets,
               int* __restrict__ cursor, int n) {
  __shared__ int sm[1024];
  int t = threadIdx.x;
  int carry = 0;
  for (int base = 0; base < n; base += 1024) {
    int idx = base + t;
    int v = (idx < n) ? deg[idx] : 0;
    sm[t] = v;
    __syncthreads();
    for (int off = 1; off < 1024; off <<= 1) {
      int add = (t >= off) ? sm[t - off] : 0;
      __syncthreads();
      sm[t] += add;
      __syncthreads();
    }
    int incl = sm[t];
    int total = sm[1023];
    if (idx < n) {
      int excl = carry + incl - v;
      offsets[idx] = excl;
      cursor[idx] = excl;
    }
    carry += total;
    __syncthreads();
  }
  if (t == 0) offsets[n] = carry;
}

// ---------------- K3: scatter src ids into CSR slots ----------------
__global__ __launch_bounds__(BLOCK)
void sage_scatter(const int* __restrict__ src, const int* __restrict__ dst,
                  int* __restrict__ cursor, int* __restrict__ srcs_sorted,
                  int n_edges) {
  int tid = blockIdx.x * blockDim.x + threadIdx.x;
  int stride = gridDim.x * blockDim.x;
  for (int e = tid; e < n_edges; e += stride) {
    int pos = atomicAdd(&cursor[dst[e]], 1);
    srcs_sorted[pos] = src[e];
  }
}

// ---------------- K4: wave-per-dst gather-sum + finalize ----------------
__global__ __launch_bounds__(BLOCK)
void sage_gather_finalize(const float* __restrict__ x,
                          const int* __restrict__ offsets,
                          const int* __restrict__ srcs,
                          const float* __restrict__ h_self,
                          const float* __restrict__ u_self,
                          float* __restrict__ out, int n_nodes) {
  const int lane = threadIdx.x & (WAVE - 1);
  const int wv = threadIdx.x >> 5;
  const int wave = blockIdx.x * WAVES_PER_BLOCK + wv;
  const int n_waves = gridDim.x * WAVES_PER_BLOCK;

#ifdef HAVE_ASYNC_LDS
  __shared__ float stage[WAVES_PER_BLOCK][2][D_FEAT];  // 8KB, double buffer/wave
#endif

  for (int d = wave; d < n_nodes; d += n_waves) {
    const int start = offsets[d];
    const int end = offsets[d + 1];
    float4 acc = {0.0f, 0.0f, 0.0f, 0.0f};

#ifdef HAVE_ASYNC_LDS
    // Double-buffered async gather: row i lands in LDS while row i-1 is summed.
    int buf = 0;
    if (start < end) {
      int s0 = srcs[start];
      async_row_16B(x + (size_t)s0 * D_FEAT + lane * 4, &stage[wv][0][lane * 4]);
    }
    for (int i = start; i < end; ++i) {
      if (i + 1 < end) {
        int sn = srcs[i + 1];
        async_row_16B(x + (size_t)sn * D_FEAT + lane * 4,
                      &stage[wv][buf ^ 1][lane * 4]);
        WAIT_ASYNC(1);  // in-order completion => row i is in LDS
      } else {
        WAIT_ASYNC(0);
      }
      const float4 v = *(const float4*)&stage[wv][buf][lane * 4];
      acc4(acc, v);
      buf ^= 1;
    }
#else
    for (int i = start; i < end; ++i) {
      int s = srcs[i];
      if (i + 1 < end) {
        int s2 = srcs[i + 1];
        __builtin_prefetch(x + (size_t)s2 * D_FEAT + lane * 4, 0, 0);
      }
      const float4 v = *(const float4*)(x + (size_t)s * D_FEAT + lane * 4);
      acc4(acc, v);
    }
#endif

    const float invd = 1.0f / fmaxf((float)(end - start), 1.0f);
    const size_t base = (size_t)d * D_FEAT + lane * 4;
    float4 h = *(const float4*)(h_self + base);
    float4 u = *(const float4*)(u_self + base);
    float4 sd = dropout4(h, u);
    float4 o;
    o.x = fmaxf(fmaf(acc.x, invd, sd.x), 0.0f);
    o.y = fmaxf(fmaf(acc.y, invd, sd.y), 0.0f);
    o.z = fmaxf(fmaf(acc.z, invd, sd.z), 0.0f);
    o.w = fmaxf(fmaf(acc.w, invd, sd.w), 0.0f);
    *(float4*)(out + base) = o;
  }
}

// ---------------- Fallback path (small workspace): atomics into d_out --------
__global__ __launch_bounds__(BLOCK)
void sage_fb_zero(float* __restrict__ out, int* __restrict__ deg,
                  int n_elems, int n_nodes) {
  int tid = blockIdx.x * blockDim.x + threadIdx.x;
  int stride = gridDim.x * blockDim.x;
  for (int i = tid; i < n_elems; i += stride) out[i] = 0.0f;
  for (int i = tid; i < n_nodes; i += stride) deg[i] = 0;
}

__global__ __launch_bounds__(BLOCK)
void sage_fb_edge(const float* __restrict__ feat, const float* __restrict__ u,
                  const int* __restrict__ src, const int* __restrict__ dst,
                  float* __restrict__ out, int* __restrict__ deg, int n_edges) {
  const int lane = threadIdx.x & (WAVE - 1);
  const int wave = (blockIdx.x * blockDim.x + threadIdx.x) >> 5;
  const int n_waves = (gridDim.x * blockDim.x) >> 5;
  for (int e = wave; e < n_edges; e += n_waves) {
    int s = src[e], d = dst[e];
    size_t sb = (size_t)s * D_FEAT + lane * 4;
    float4 f = *(const float4*)(feat + sb);
    float4 uu = *(const float4*)(u + sb);
    float4 xv = dropout4(f, uu);
    float* o = out + (size_t)d * D_FEAT + lane * 4;
    unsafeAtomicAdd(o + 0, xv.x);
    unsafeAtomicAdd(o + 1, xv.y);
    unsafeAtomicAdd(o + 2, xv.z);
    unsafeAtomicAdd(o + 3, xv.w);
    if (lane == 0) atomicAdd(&deg[d], 1);
  }
}

__global__ __launch_bounds__(BLOCK)
void sage_fb_finalize(float* __restrict__ out, const int* __restrict__ deg,
                      const float* __restrict__ h_self,
                      const float* __restrict__ u_self, int n_nodes) {
  const int lane = threadIdx.x & (WAVE - 1);
  const int wave = (blockIdx.x * blockDim.x + threadIdx.x) >> 5;
  const int n_waves = (gridDim.x * blockDim.x) >> 5;
  for (int d = wave; d < n_nodes; d += n_waves) {
    const float invd = 1.0f / fmaxf((float)deg[d], 1.0f);
    const size_t base = (size_t)d * D_FEAT + lane * 4;
    float4 a = *(const float4*)(out + base);
    float4 h = *(const float4*)(h_self + base);
    float4 u = *(const float4*)(u_self + base);
    float4 sd = dropout4(h, u);
    float4 o;
    o.x = fmaxf(fmaf(a.x, invd, sd.x), 0.0f);
    o.y = fmaxf(fmaf(a.y, invd, sd.y), 0.0f);
    o.z = fmaxf(fmaf(a.z, invd, sd.z), 0.0f);
    o.w = fmaxf(fmaf(a.w, invd, sd.w), 0.0f);
    *(float4*)(out + base) = o;
  }
}

extern "C" void kernel_launch(void* const* d_in, const int* in_sizes, int n_in,
                              void* d_out, int out_size, void* d_ws, size_t ws_size,
                              hipStream_t stream) {
  const float* feat_src = (const float*)d_in[0];
  const float* h_self   = (const float*)d_in[1];
  const float* u_src    = (const float*)d_in[2];
  const float* u_self   = (const float*)d_in[3];
  const int*   src      = (const int*)d_in[4];
  const int*   dst      = (const int*)d_in[5];
  float* out = (float*)d_out;

  const int n_nodes = in_sizes[0] / D_FEAT;
  const int n_edges = in_sizes[4];
  const size_t nd = (size_t)n_nodes * D_FEAT;

  const size_t need_csr =
      nd * 4 + (size_t)n_nodes * 4 + (size_t)(n_nodes + 1) * 4 +
      (size_t)n_nodes * 4 + (size_t)n_edges * 4;

  if (ws_size >= need_csr) {
    float* x = (float*)d_ws;
    int* deg = (int*)(x + nd);
    int* offsets = deg + n_nodes;          // n_nodes + 1 entries
    int* cursor = offsets + n_nodes + 1;
    int* srcs_sorted = cursor + n_nodes;

    sage_init_x_deg<<<2048, BLOCK, 0, stream>>>(feat_src, u_src, x, deg,
                                                (int)(nd / 4), n_nodes);
    sage_hist<<<1280, BLOCK, 0, stream>>>(dst, deg, n_edges);
    sage_scan<<<1, 1024, 0, stream>>>(deg, offsets, cursor, n_nodes);
    sage_scatter<<<1280, BLOCK, 0, stream>>>(src, dst, cursor, srcs_sorted,
                                             n_edges);
    int gblocks = (n_nodes + WAVES_PER_BLOCK - 1) / WAVES_PER_BLOCK;
    sage_gather_finalize<<<gblocks, BLOCK, 0, stream>>>(
        x, offsets, srcs_sorted, h_self, u_self, out, n_nodes);
  } else {
    // Minimal-workspace fallback: aggregate with f32 atomics into d_out.
    int* deg = (int*)d_ws;  // needs n_nodes * 4 bytes
    sage_fb_zero<<<2048, BLOCK, 0, stream>>>(out, deg, (int)nd, n_nodes);
    sage_fb_edge<<<2048, BLOCK, 0, stream>>>(feat_src, u_src, src, dst, out,
                                             deg, n_edges);
    sage_fb_finalize<<<1024, BLOCK, 0, stream>>>(out, deg, h_self, u_self,
                                                 n_nodes);
  }
}